// TrilinearInterpolation_59906203845136
// MI455X (gfx1250) — compile-verified
//
#include <hip/hip_runtime.h>
#include <stdint.h>

// Problem constants (fixed by setup_inputs)
#define B_   4
#define F_   8
#define D_   128
#define H_   128
#define W_   128
#define HG_  512
#define WG_  512
#define YC_  4                 // y-rows per TDM tile: tile = [F=8][YC=4][W=128] f32 = 16 KB LDS
#define PLANE_ ((size_t)HG_ * WG_)
#define DHW_   ((size_t)D_ * H_ * W_)

typedef unsigned int u32x4 __attribute__((ext_vector_type(4)));
typedef unsigned int u32x8 __attribute__((ext_vector_type(8)));

// ---------------------------------------------------------------------------
// Phase 1: [B,F,D,H,W] -> [B,D,H,W,F] using the Tensor Data Mover.
// One TDM descriptor per block pulls a 3D tile (dim0=x contiguous, dim1=y
// stride W, dim2=f stride D*H*W) into contiguous LDS; all 8 waves then do a
// coalesced float4 writeout in channels-last order. Writeout stays RT so the
// channels-last volume lands in L2 for the gather pass.
// ---------------------------------------------------------------------------
__global__ void __launch_bounds__(256)
tdm_transpose_cl(const float* __restrict__ in, float* __restrict__ outv) {
  __shared__ float tile[F_ * YC_ * W_];   // 4096 floats = 16 KB

  int t = blockIdx.x;
  const int yc = t % (H_ / YC_); t /= (H_ / YC_);
  const int z  = t % D_;
  const int b  = t / D_;
  const int y0 = yc * YC_;

  if (threadIdx.x < 32u) {  // wave 0 issues the DMA (EXEC is ignored by TDM)
    const uint64_t gaddr = (uint64_t)(uintptr_t)in +
        4ull * ((((uint64_t)(b * F_) * D_ + (uint64_t)z) * H_ + (uint64_t)y0) * W_);
    const uint32_t lds_base = (uint32_t)(uintptr_t)(&tile[0]);   // low 32 bits = LDS byte offset

    // ---- D# group 0 (ISA 08 §8.3): count=1 | lds_addr | global_addr | type=2
    u32x4 g0;
    g0.x = 1u;
    g0.y = lds_base;
    g0.z = (uint32_t)gaddr;
    g0.w = ((uint32_t)(gaddr >> 32) & 0x01FFFFFFu) | (2u << 30);

    // ---- D# group 1 (§8.4)
    u32x8 g1;
    g1.s0 = (2u << 16);                              // data_size=2 (4 bytes); mask=0; no pad/iter
    g1.s1 = ((uint32_t)W_  << 16);                   // [63:48]  tensor_dim0 lo16 = 128
    g1.s2 = ((uint32_t)YC_ << 16);                   // [95:80]  tensor_dim1 lo16 = YC (tile-local)
    g1.s3 = ((uint32_t)W_  << 16);                   // [127:112] tile_dim0 = 128
    g1.s4 = (uint32_t)YC_ | ((uint32_t)F_ << 16);    // tile_dim1=YC, tile_dim2=F
    g1.s5 = (uint32_t)W_;                            // tensor_dim0_stride = W   (y-line stride)
    g1.s6 = ((uint32_t)DHW_ & 0xFFFFu) << 16;        // tensor_dim1_stride[15:0] (f-plane stride lo)
    g1.s7 = (uint32_t)((uint64_t)DHW_ >> 16);        // tensor_dim1_stride[47:16]

    // ---- D# group 2/3 (§8.5/8.6): tensor_dim2=F, rest unused
    u32x4 g2; g2.x = (uint32_t)F_; g2.y = 0u; g2.z = 0u; g2.w = 0u;
    u32x4 g3; g3.x = 0u; g3.y = 0u; g3.z = 0u; g3.w = 0u;

    asm volatile("tensor_load_to_lds %0, %1, %2, %3"
                 :: "s"(g0), "s"(g1), "s"(g2), "s"(g3)
                 : "memory");
    __builtin_amdgcn_s_wait_tensorcnt(0);
  }
  __syncthreads();

  // Writeout: channels-last [YC][W][F] as lane-consecutive float4s (fully coalesced).
  float* outBase = outv + (((size_t)b * D_ + z) * H_ + y0) * (size_t)(W_ * F_);
  const int nq = (YC_ * W_ * F_) / 4;                 // 1024 float4s
  for (int q = (int)threadIdx.x; q < nq; q += (int)blockDim.x) {
    const int e  = q * 4;
    const int f0 = e % F_;                            // 0 or 4
    const int x  = (e / F_) % W_;
    const int yl = e / (F_ * W_);
    float4 vv;
    vv.x = tile[(f0 + 0) * (YC_ * W_) + yl * W_ + x];
    vv.y = tile[(f0 + 1) * (YC_ * W_) + yl * W_ + x];
    vv.z = tile[(f0 + 2) * (YC_ * W_) + yl * W_ + x];
    vv.w = tile[(f0 + 3) * (YC_ * W_) + yl * W_ + x];
    ((float4*)outBase)[q] = vv;
  }
}

// ---------------------------------------------------------------------------
// Phase 2: trilinear gather from channels-last volume.
// Per point: 8 corners x 2x float4 (each corner's 8 channels are 32B
// contiguous; the x0/x1 pair shares a 64B line) -> 16 global_load_b128.
// Grid reads and output stores are non-temporal: the only data worth keeping
// in the 192 MB L2 is the 256 MB channels-last volume.
// ---------------------------------------------------------------------------
__device__ __forceinline__ int clampi_f(float a, int hi) {
  return (int)fminf(fmaxf(a, 0.0f), (float)hi);
}

__global__ void __launch_bounds__(256)
trilerp_cl(const float* __restrict__ volcl, const float* __restrict__ grid,
           float* __restrict__ out) {
  const int p = blockIdx.x * blockDim.x + threadIdx.x;
  const int b = p / (HG_ * WG_);
  const int r = p - b * (HG_ * WG_);

  const float* gp = grid + 3 * (size_t)p;
  const float gx = fminf(fmaxf(__builtin_nontemporal_load(gp + 0), -1.0f), 1.0f);
  const float gy = fminf(fmaxf(__builtin_nontemporal_load(gp + 1), -1.0f), 1.0f);
  const float gz = fminf(fmaxf(__builtin_nontemporal_load(gp + 2), -1.0f), 1.0f);

  const float x  = (gx + 1.0f) * 0.5f * (float)(W_ - 1);
  const float y  = (gy + 1.0f) * 0.5f * (float)(H_ - 1);
  const float zc = (gz + 1.0f) * 0.5f * (float)(D_ - 1);

  const float xf = floorf(x), yf = floorf(y), zf = floorf(zc);
  const float u = x - xf, v = y - yf, w = zc - zf;

  const int ix0 = clampi_f(xf, W_ - 1), ix1 = clampi_f(xf + 1.0f, W_ - 1);
  const int iy0 = clampi_f(yf, H_ - 1), iy1 = clampi_f(yf + 1.0f, H_ - 1);
  const int iz0 = clampi_f(zf, D_ - 1), iz1 = clampi_f(zf + 1.0f, D_ - 1);

  const float u0 = 1.0f - u, v0 = 1.0f - v, w0 = 1.0f - w;
  const float w000 = u0 * v0 * w0, w001 = u0 * v0 * w;
  const float w010 = u0 * v  * w0, w011 = u0 * v  * w;
  const float w100 = u  * v0 * w0, w101 = u  * v0 * w;
  const float w110 = u  * v  * w0, w111 = u  * v  * w;

  const float* vol = volcl + (size_t)b * (DHW_ * F_);
  float4 aLo = {0.f, 0.f, 0.f, 0.f}, aHi = {0.f, 0.f, 0.f, 0.f};

  auto acc = [&](float wt, int iz, int iy, int ix) {
    const float4* c = (const float4*)(vol + ((((size_t)iz) * H_ + iy) * W_ + ix) * F_);
    const float4 lo = c[0], hi = c[1];
    aLo.x = fmaf(wt, lo.x, aLo.x); aLo.y = fmaf(wt, lo.y, aLo.y);
    aLo.z = fmaf(wt, lo.z, aLo.z); aLo.w = fmaf(wt, lo.w, aLo.w);
    aHi.x = fmaf(wt, hi.x, aHi.x); aHi.y = fmaf(wt, hi.y, aHi.y);
    aHi.z = fmaf(wt, hi.z, aHi.z); aHi.w = fmaf(wt, hi.w, aHi.w);
  };

  acc(w000, iz0, iy0, ix0); acc(w001, iz1, iy0, ix0);
  acc(w010, iz0, iy1, ix0); acc(w011, iz1, iy1, ix0);
  acc(w100, iz0, iy0, ix1); acc(w101, iz1, iy0, ix1);
  acc(w110, iz0, iy1, ix1); acc(w111, iz1, iy1, ix1);

  float* ob = out + (size_t)b * F_ * PLANE_ + (size_t)r;
  __builtin_nontemporal_store(aLo.x, ob + 0 * PLANE_);
  __builtin_nontemporal_store(aLo.y, ob + 1 * PLANE_);
  __builtin_nontemporal_store(aLo.z, ob + 2 * PLANE_);
  __builtin_nontemporal_store(aLo.w, ob + 3 * PLANE_);
  __builtin_nontemporal_store(aHi.x, ob + 4 * PLANE_);
  __builtin_nontemporal_store(aHi.y, ob + 5 * PLANE_);
  __builtin_nontemporal_store(aHi.z, ob + 6 * PLANE_);
  __builtin_nontemporal_store(aHi.w, ob + 7 * PLANE_);
}

// ---------------------------------------------------------------------------
// Fallback: direct gather from [B,F,D,H,W] if workspace is too small.
// ---------------------------------------------------------------------------
__global__ void __launch_bounds__(256)
trilerp_direct(const float* __restrict__ feats, const float* __restrict__ grid,
               float* __restrict__ out) {
  const int p = blockIdx.x * blockDim.x + threadIdx.x;
  const int b = p / (HG_ * WG_);
  const int r = p - b * (HG_ * WG_);

  const float* gp = grid + 3 * (size_t)p;
  const float gx = fminf(fmaxf(__builtin_nontemporal_load(gp + 0), -1.0f), 1.0f);
  const float gy = fminf(fmaxf(__builtin_nontemporal_load(gp + 1), -1.0f), 1.0f);
  const float gz = fminf(fmaxf(__builtin_nontemporal_load(gp + 2), -1.0f), 1.0f);

  const float x  = (gx + 1.0f) * 0.5f * (float)(W_ - 1);
  const float y  = (gy + 1.0f) * 0.5f * (float)(H_ - 1);
  const float zc = (gz + 1.0f) * 0.5f * (float)(D_ - 1);

  const float xf = floorf(x), yf = floorf(y), zf = floorf(zc);
  const float u = x - xf, v = y - yf, w = zc - zf;

  const int ix0 = clampi_f(xf, W_ - 1), ix1 = clampi_f(xf + 1.0f, W_ - 1);
  const int iy0 = clampi_f(yf, H_ - 1), iy1 = clampi_f(yf + 1.0f, H_ - 1);
  const int iz0 = clampi_f(zf, D_ - 1), iz1 = clampi_f(zf + 1.0f, D_ - 1);

  const float u0 = 1.0f - u, v0 = 1.0f - v, w0 = 1.0f - w;
  const float wt[8] = {u0 * v0 * w0, u0 * v0 * w, u0 * v * w0, u0 * v * w,
                       u  * v0 * w0, u  * v0 * w, u  * v * w0, u  * v * w};
  const size_t off[8] = {
      ((size_t)iz0 * H_ + iy0) * W_ + ix0, ((size_t)iz1 * H_ + iy0) * W_ + ix0,
      ((size_t)iz0 * H_ + iy1) * W_ + ix0, ((size_t)iz1 * H_ + iy1) * W_ + ix0,
      ((size_t)iz0 * H_ + iy0) * W_ + ix1, ((size_t)iz1 * H_ + iy0) * W_ + ix1,
      ((size_t)iz0 * H_ + iy1) * W_ + ix1, ((size_t)iz1 * H_ + iy1) * W_ + ix1};

  const float* volb = feats + (size_t)b * F_ * DHW_;
  float* ob = out + (size_t)b * F_ * PLANE_ + (size_t)r;
#pragma unroll
  for (int f = 0; f < F_; ++f) {
    const float* base = volb + (size_t)f * DHW_;
    float a = 0.0f;
#pragma unroll
    for (int c = 0; c < 8; ++c) a = fmaf(wt[c], base[off[c]], a);
    __builtin_nontemporal_store(a, ob + (size_t)f * PLANE_);
  }
}

// ---------------------------------------------------------------------------
extern "C" void kernel_launch(void* const* d_in, const int* in_sizes, int n_in,
                              void* d_out, int out_size, void* d_ws, size_t ws_size,
                              hipStream_t stream) {
  (void)in_sizes; (void)n_in; (void)out_size;
  const float* feats = (const float*)d_in[0];
  const float* grid  = (const float*)d_in[1];
  float* out = (float*)d_out;

  const size_t need = (size_t)B_ * DHW_ * F_ * sizeof(float);  // 256 MB channels-last copy
  const int np = B_ * HG_ * WG_;                                // 1,048,576 points

  if (ws_size >= need) {
    float* volcl = (float*)d_ws;
    const int nblk1 = B_ * D_ * (H_ / YC_);                     // 16384 tiles
    tdm_transpose_cl<<<nblk1, 256, 0, stream>>>(feats, volcl);
    trilerp_cl<<<np / 256, 256, 0, stream>>>(volcl, grid, out);
  } else {
    trilerp_direct<<<np / 256, 256, 0, stream>>>(feats, grid, out);
  }
}